// DGCNN_Block_28028956574155
// MI455X (gfx1250) — compile-verified
//
#include <hip/hip_runtime.h>

typedef __attribute__((ext_vector_type(2))) float v2f;
typedef __attribute__((ext_vector_type(8))) float v8f;

#define BATCH 8
#define CCH   128
#define NPTS  4096
#define KNN   9

static __device__ __forceinline__ v8f wmma_f32(v2f a, v2f b, v8f c) {
  // D = A(16x4, f32) x B(4x16, f32) + C(16x16, f32)
  return __builtin_amdgcn_wmma_f32_16x16x4_f32(false, a, false, b, (short)0, c, false, false);
}

// LDS byte offset of a __shared__ object (low 32 bits of the flat LDS pointer)
static __device__ __forceinline__ unsigned ldsoff(const void* p) {
  return (unsigned)(unsigned long long)(uintptr_t)p;
}

static __device__ __forceinline__ void wait_async0() {
#if __has_builtin(__builtin_amdgcn_s_wait_asynccnt)
  __builtin_amdgcn_s_wait_asynccnt(0);
#else
  asm volatile("s_wait_asynccnt 0x0" ::: "memory");
#endif
}

// async memory -> LDS, 4 bytes, GVS addressing (SGPR64 base + VGPR32 byte offset)
static __device__ __forceinline__ void async_ld_b32(unsigned lds_b, unsigned goff_b,
                                                    const void* base) {
  asm volatile("global_load_async_to_lds_b32 %0, %1, %2"
               :: "v"(lds_b), "v"(goff_b),
                  "s"((unsigned long long)(uintptr_t)base)
               : "memory");
}

// async memory -> LDS, 16 bytes
static __device__ __forceinline__ void async_ld_b128(unsigned lds_b, unsigned goff_b,
                                                     const void* base) {
  asm volatile("global_load_async_to_lds_b128 %0, %1, %2"
               :: "v"(lds_b), "v"(goff_b),
                  "s"((unsigned long long)(uintptr_t)base)
               : "memory");
}

// ---------------------------------------------------------------- sq norms
__global__ __launch_bounds__(256) void sq_kernel(const float* __restrict__ x,
                                                 float* __restrict__ sq) {
  int i = blockIdx.x * 256 + threadIdx.x;           // over B*N
  if (i >= BATCH * NPTS) return;
  int b = i >> 12, n = i & (NPTS - 1);
  const float* xb = x + (size_t)b * CCH * NPTS + n;
  float s = 0.f;
  #pragma unroll 8
  for (int c = 0; c < CCH; ++c) { float v = xb[c * NPTS]; s += v * v; }
  sq[i] = s;
}

// ---------------------------------------------------------------- kNN (WMMA pd + top-9)
__global__ __launch_bounds__(32) void knn_kernel(const float* __restrict__ x,
                                                 const float* __restrict__ sq,
                                                 int* __restrict__ idx_out) {
  const int b  = blockIdx.y;
  const int n0 = blockIdx.x * 16;
  const int lane = threadIdx.x;
  const int l15 = lane & 15, lhi = lane >> 4;
  __shared__ __align__(16) float As[CCH][16];
  __shared__ float sqn[16];
  __shared__ float tile[16][17];
  const float* xb = x + (size_t)b * CCH * NPTS;

  // A strip (128x16, contiguous 16-float rows): async b128 direct to LDS
  {
    unsigned asb = ldsoff(&As[0][0]);
    for (int e = lane; e < CCH * 4; e += 32) {        // 512 float4 chunks
      int k = e >> 2, i0 = (e & 3) * 4;
      async_ld_b128(asb + (unsigned)e * 16u,
                    (unsigned)((k * NPTS + n0 + i0) * 4), xb);
    }
  }
  if (lane < 16) sqn[lane] = sq[b * NPTS + n0 + lane];
  wait_async0();
  __syncthreads();

  float bestv[KNN]; int besti[KNN];
  #pragma unroll
  for (int p = 0; p < KNN; ++p) { bestv[p] = -3.0e38f; besti[p] = 0; }

  for (int m0 = 0; m0 < NPTS; m0 += 16) {
    if (m0 + 16 < NPTS) {
      #pragma unroll
      for (int q = 0; q < 4; ++q)
        __builtin_prefetch(&xb[(lane + 32 * q) * NPTS + m0 + 16], 0, 3);
    }
    v8f acc = {};
    #pragma unroll 4
    for (int k0 = 0; k0 < CCH; k0 += 4) {
      int ka = k0 + lhi * 2;
      v2f a, bb;
      a.x  = As[ka][l15];     a.y  = As[ka + 1][l15];
      bb.x = xb[ka * NPTS + m0 + l15];
      bb.y = xb[(ka + 1) * NPTS + m0 + l15];
      acc = wmma_f32(a, bb, acc);
    }
    float sqm = sq[b * NPTS + m0 + l15];
    #pragma unroll
    for (int v = 0; v < 8; ++v) {
      int row = v + 8 * lhi;
      tile[row][l15] = 2.0f * acc[v] - sqn[row] - sqm;   // pd = -||xi-xj||^2
    }
    __syncthreads();
    if (lane < 16) {
      #pragma unroll
      for (int j = 0; j < 16; ++j) {
        float val = tile[lane][j];
        if (val > bestv[KNN - 1]) {
          bestv[KNN - 1] = val; besti[KNN - 1] = m0 + j;
          #pragma unroll
          for (int p = KNN - 1; p > 0; --p) {
            if (bestv[p] > bestv[p - 1]) {
              float tv = bestv[p]; bestv[p] = bestv[p - 1]; bestv[p - 1] = tv;
              int   ti = besti[p]; besti[p] = besti[p - 1]; besti[p - 1] = ti;
            }
          }
        }
      }
    }
    __syncthreads();
  }
  if (lane < 16) {
    int* o = idx_out + ((size_t)b * NPTS + n0 + lane) * KNN;
    #pragma unroll
    for (int p = 0; p < KNN; ++p) o[p] = besti[p];
  }
}

// ---------------------------------------------------------------- weight prep (transposed [k][o])
__global__ __launch_bounds__(256) void prep_kernel(const float* __restrict__ w1,
                                                   const float* __restrict__ w2,
                                                   float* __restrict__ Actr,
                                                   float* __restrict__ Aneg,
                                                   float* __restrict__ w2p) {
  int i = blockIdx.x * 256 + threadIdx.x;
  if (i < CCH * CCH) {                       // Actr[c][o] = sum_t (w1[o,c,t] + w1[o,c+128,t])
    int c = i >> 7, o = i & 127;
    float s = 0.f;
    #pragma unroll
    for (int t = 0; t < 3; ++t) s += w1[o * 768 + c * 3 + t] + w1[o * 768 + (c + 128) * 3 + t];
    Actr[c * CCH + o] = s;
  }
  if (i < 384 * CCH) {                       // kk = t*128 + c
    int kk = i >> 7, o = i & 127;
    int t = kk >> 7, c = kk & 127;
    Aneg[kk * CCH + o] = -w1[o * 768 + (c + 128) * 3 + t];
    w2p[kk * CCH + o]  =  w2[o * 384 + c * 3 + t];
  }
}

// ---------------------------------------------------------------- conv1 (async gather + WMMA)
__global__ __launch_bounds__(256) void conv1_kernel(const float* __restrict__ x,
                                                    const int* __restrict__ idx,
                                                    const float* __restrict__ Actr,
                                                    const float* __restrict__ Aneg,
                                                    const float* __restrict__ b1,
                                                    float* __restrict__ hpre) {
  const int b  = blockIdx.y;
  const int n0 = blockIdx.x * 16;
  const int tid = threadIdx.x;
  const int w = tid >> 5, lane = tid & 31;
  const int l15 = lane & 15, lhi = lane >> 4;
  __shared__ __align__(16) float Bc[CCH][20];   // 16B-aligned rows, conflict-free
  __shared__ float Bs[CCH][49];
  __shared__ int   idxL[16 * KNN];
  const float* xb = x + (size_t)b * CCH * NPTS;

  for (int e = tid; e < 16 * KNN; e += 256)
    idxL[e] = idx[((size_t)b * NPTS + n0 + (e / KNN)) * KNN + (e % KNN)];
  {
    unsigned bcb = ldsoff(&Bc[0][0]);
    for (int e = tid; e < CCH * 4; e += 256) {       // center strip via async b128
      int c = e >> 2, i0 = (e & 3) * 4;
      async_ld_b128(bcb + (unsigned)((c * 20 + i0) * 4),
                    (unsigned)((c * NPTS + n0 + i0) * 4), xb);
    }
  }
  wait_async0();
  __syncthreads();

  v8f acc0, acc1, acc2;
  {
    v8f c = {};
    #pragma unroll 4
    for (int k0 = 0; k0 < CCH; k0 += 4) {
      int ka = k0 + lhi * 2;
      v2f a, bb;
      a.x  = Actr[ka * CCH + 16 * w + l15];
      a.y  = Actr[(ka + 1) * CCH + 16 * w + l15];
      bb.x = Bc[ka][l15]; bb.y = Bc[ka + 1][l15];
      c = wmma_f32(a, bb, c);
    }
    acc0 = c; acc1 = c; acc2 = c;            // center term is j-independent
  }

  unsigned bsb = ldsoff(&Bs[0][0]);
  for (int t = 0; t < 3; ++t) {
    __syncthreads();                          // all waves done reading previous Bs
    for (int e = tid; e < CCH * 48; e += 256) {   // Bs[c][j*16+ni] = x[c, idx[n,3j+t]]
      int c = e / 48, col = e % 48;
      int j = col >> 4, ni = col & 15;
      int p = idxL[ni * KNN + 3 * j + t];
      async_ld_b32(bsb + (unsigned)((c * 49 + col) * 4),
                   (unsigned)((c * NPTS + p) * 4), xb);
    }
    wait_async0();
    __syncthreads();
    const float* At = Aneg + t * CCH * CCH;
    #pragma unroll 2
    for (int k0 = 0; k0 < CCH; k0 += 4) {
      int ka = k0 + lhi * 2;
      v2f a, fb0, fb1, fb2;
      a.x = At[ka * CCH + 16 * w + l15];
      a.y = At[(ka + 1) * CCH + 16 * w + l15];
      fb0.x = Bs[ka][l15];      fb0.y = Bs[ka + 1][l15];
      fb1.x = Bs[ka][16 + l15]; fb1.y = Bs[ka + 1][16 + l15];
      fb2.x = Bs[ka][32 + l15]; fb2.y = Bs[ka + 1][32 + l15];
      acc0 = wmma_f32(a, fb0, acc0);
      acc1 = wmma_f32(a, fb1, acc1);
      acc2 = wmma_f32(a, fb2, acc2);
    }
  }

  // store hpre[(b*N+n)*384 + j*128 + o] (+ bias)
  int ob = 16 * w + 8 * lhi;
  float4 blo = *(const float4*)(b1 + ob);
  float4 bhi = *(const float4*)(b1 + ob + 4);
  size_t nb = ((size_t)(b * NPTS + n0 + l15)) * 384 + ob;
  v8f* accs[3] = { &acc0, &acc1, &acc2 };
  #pragma unroll
  for (int j = 0; j < 3; ++j) {
    v8f a = *accs[j];
    float4 lo = make_float4(a[0] + blo.x, a[1] + blo.y, a[2] + blo.z, a[3] + blo.w);
    float4 hi = make_float4(a[4] + bhi.x, a[5] + bhi.y, a[6] + bhi.z, a[7] + bhi.w);
    *(float4*)(hpre + nb + (size_t)j * 128)     = lo;
    *(float4*)(hpre + nb + (size_t)j * 128 + 4) = hi;
  }
}

// ---------------------------------------------------------------- BN stats over hpre (per channel c = k%128)
__global__ __launch_bounds__(128) void stats1_kernel(const float* __restrict__ hpre,
                                                     float* __restrict__ sums) {
  int c = threadIdx.x;
  size_t r0 = (size_t)blockIdx.x * 384;      // 256 blocks * 384 rows = B*N*3
  float s = 0.f, s2 = 0.f;
  for (int r = 0; r < 384; ++r) {
    float v = hpre[(r0 + r) * 128 + c];
    s += v; s2 += v * v;
  }
  atomicAdd(&sums[c], s);
  atomicAdd(&sums[128 + c], s2);
}

__global__ __launch_bounds__(128) void bnfin_kernel(const float* __restrict__ sums,
                                                    const float* __restrict__ g,
                                                    const float* __restrict__ beta,
                                                    float cnt,
                                                    float* __restrict__ scale,
                                                    float* __restrict__ shift) {
  int c = threadIdx.x;
  float mean = sums[c] / cnt;
  float var  = sums[128 + c] / cnt - mean * mean;
  float sc   = g[c] * rsqrtf(var + 1e-5f);
  scale[c] = sc;
  shift[c] = beta[c] - mean * sc;
}

// ---------------------------------------------------------------- conv2 (BN1+ReLU fused into B build, K=384)
__global__ __launch_bounds__(256) void conv2_kernel(const float* __restrict__ hpre,
                                                    const float* __restrict__ w2p,
                                                    const float* __restrict__ scale1,
                                                    const float* __restrict__ shift1,
                                                    const float* __restrict__ b2,
                                                    float* __restrict__ outpre) {
  const int b  = blockIdx.y;
  const int n0 = blockIdx.x * 16;
  const int tid = threadIdx.x;
  const int w = tid >> 5, lane = tid & 31;
  const int l15 = lane & 15, lhi = lane >> 4;
  __shared__ float Bs[384][17];
  __shared__ float sc[128], sh[128];
  if (tid < 128) { sc[tid] = scale1[tid]; sh[tid] = shift1[tid]; }
  __syncthreads();
  for (int e = tid; e < 384 * 16; e += 256) {     // contiguous per column
    int ni = e / 384, k = e % 384;
    float v = hpre[(size_t)(b * NPTS + n0 + ni) * 384 + k];
    int c = k & 127;
    v = v * sc[c] + sh[c];
    Bs[k][ni] = v > 0.f ? v : 0.f;
  }
  __syncthreads();

  v8f acc = {};
  #pragma unroll 4
  for (int k0 = 0; k0 < 384; k0 += 4) {
    int ka = k0 + lhi * 2;
    v2f a, bb;
    a.x  = w2p[ka * CCH + 16 * w + l15];
    a.y  = w2p[(ka + 1) * CCH + 16 * w + l15];
    bb.x = Bs[ka][l15]; bb.y = Bs[ka + 1][l15];
    acc = wmma_f32(a, bb, acc);
  }
  #pragma unroll
  for (int v = 0; v < 8; ++v) {
    int o = 16 * w + 8 * lhi + v;
    outpre[((size_t)(b * CCH + o)) * NPTS + n0 + l15] = acc[v] + b2[o];
  }
}

// ---------------------------------------------------------------- BN stats over outpre (per channel o)
__global__ __launch_bounds__(256) void stats2_kernel(const float* __restrict__ outpre,
                                                     float* __restrict__ sums) {
  int o = blockIdx.x & 127;                  // layout (b*128+o)*4096
  const float* p = outpre + (size_t)blockIdx.x * NPTS;
  float s = 0.f, s2 = 0.f;
  for (int i = threadIdx.x; i < NPTS; i += 256) { float v = p[i]; s += v; s2 += v * v; }
  __shared__ float rs[256], rs2[256];
  rs[threadIdx.x] = s; rs2[threadIdx.x] = s2;
  __syncthreads();
  for (int st = 128; st > 0; st >>= 1) {
    if (threadIdx.x < st) { rs[threadIdx.x] += rs[threadIdx.x + st]; rs2[threadIdx.x] += rs2[threadIdx.x + st]; }
    __syncthreads();
  }
  if (threadIdx.x == 0) { atomicAdd(&sums[o], rs[0]); atomicAdd(&sums[128 + o], rs2[0]); }
}

// ---------------------------------------------------------------- finalize BN2+ReLU
__global__ __launch_bounds__(256) void final_kernel(const float* __restrict__ outpre,
                                                    const float* __restrict__ scale,
                                                    const float* __restrict__ shift,
                                                    float* __restrict__ out) {
  size_t i = (size_t)blockIdx.x * 256 + threadIdx.x;  // 8*128*4096
  int o = (int)((i >> 12) & 127);
  float v = outpre[i] * scale[o] + shift[o];
  out[i] = v > 0.f ? v : 0.f;
}

// ---------------------------------------------------------------- host launcher
extern "C" void kernel_launch(void* const* d_in, const int* in_sizes, int n_in,
                              void* d_out, int out_size, void* d_ws, size_t ws_size,
                              hipStream_t stream) {
  (void)in_sizes; (void)n_in; (void)out_size; (void)ws_size;
  const float* features = (const float*)d_in[0];
  const float* w1    = (const float*)d_in[1];
  const float* b1    = (const float*)d_in[2];
  const float* g1    = (const float*)d_in[3];
  const float* beta1 = (const float*)d_in[4];
  const float* w2    = (const float*)d_in[5];
  const float* b2    = (const float*)d_in[6];
  const float* g2    = (const float*)d_in[7];
  const float* beta2 = (const float*)d_in[8];
  float* out = (float*)d_out;

  float* f     = (float*)d_ws;
  float* sq    = f;                                   // B*N
  int*   idxp  = (int*)(f + BATCH * NPTS);            // B*N*9
  float* Actr  = f + BATCH * NPTS + BATCH * NPTS * KNN;
  float* Aneg  = Actr + CCH * CCH;                    // 384*128
  float* w2p   = Aneg + 384 * CCH;                    // 384*128
  float* hpre  = w2p + 384 * CCH;                     // B*N*3*128
  float* outp  = hpre + (size_t)BATCH * NPTS * 3 * CCH; // B*128*N
  float* st1   = outp + (size_t)BATCH * CCH * NPTS;   // 256
  float* sc1   = st1 + 256;
  float* sh1   = sc1 + 128;
  float* st2   = sh1 + 128;                           // 256
  float* sc2   = st2 + 256;
  float* sh2   = sc2 + 128;

  hipMemsetAsync(st1, 0, 1024 * sizeof(float), stream);  // zeros st1..sh2

  sq_kernel  <<<dim3(BATCH * NPTS / 256), 256, 0, stream>>>(features, sq);
  knn_kernel <<<dim3(NPTS / 16, BATCH), 32, 0, stream>>>(features, sq, idxp);
  prep_kernel<<<dim3(192), 256, 0, stream>>>(w1, w2, Actr, Aneg, w2p);
  conv1_kernel<<<dim3(NPTS / 16, BATCH), 256, 0, stream>>>(features, idxp, Actr, Aneg, b1, hpre);
  stats1_kernel<<<dim3(256), 128, 0, stream>>>(hpre, st1);
  bnfin_kernel<<<dim3(1), 128, 0, stream>>>(st1, g1, beta1, (float)(BATCH * NPTS * 3), sc1, sh1);
  conv2_kernel<<<dim3(NPTS / 16, BATCH), 256, 0, stream>>>(hpre, w2p, sc1, sh1, b2, outp);
  stats2_kernel<<<dim3(BATCH * CCH), 256, 0, stream>>>(outp, st2);
  bnfin_kernel<<<dim3(1), 128, 0, stream>>>(st2, g2, beta2, (float)(BATCH * NPTS), sc2, sh2);
  final_kernel<<<dim3(BATCH * CCH * NPTS / 256), 256, 0, stream>>>(outp, sc2, sh2, out);
}